// ConvCNP_Multi_CC_54881092108534
// MI455X (gfx1250) — compile-verified
//
#include <hip/hip_runtime.h>
#include <math.h>

#define NBATCH 4
#define NC 512
#define NG 2048
#define CH 3
#define NPAR 3
#define OUTC 64
#define NBAS 10
#define EPSV 1e-6f
#define BN_EPSV 1e-5f
#define KDIM 18          // 2*ch*npar
#define KPAD 20          // pad K to multiple of 4 for WMMA f32 16x16x4
#define MROWS (NBATCH*NG) // 8192

// workspace layout (float offsets)
#define WS_FEAT 0
#define WS_PRE  (MROWS*KPAD)                    // conv input, (p,c,b,g)
#define WS_Z    (WS_PRE + NPAR*CH*NBATCH*NG)    // conv output, (p,c,b,g)
#define WS_STAT (WS_Z + NPAR*CH*NBATCH*NG)      // 9 x {sum, sumsq} (64 slots)
#define WS_GWP  (WS_STAT + 64)                  // zero-padded g_w, (OUTC, KPAD)

// output layout (float offsets): y_out | n_f | fourier_prior | n_h1 | h0_f
#define O_YOUT 0
#define O_NF   (MROWS*OUTC)          // 524288
#define O_FP   (O_NF + MROWS*9)      // 598016
#define O_NH1  (O_FP + MROWS*9)      // 671744
#define O_H0F  (O_NH1 + MROWS*9)     // 745472

typedef __attribute__((ext_vector_type(2))) float v2f;
typedef __attribute__((ext_vector_type(8))) float v8f;

// Setup: zero atomic stats + build zero-padded g_w (OUTC x KPAD) so the GEMM's
// B-fragment loads are unconditional global_load_b64 (no EXEC-masked tails).
__global__ void k_setup(const float* __restrict__ gw, float* __restrict__ ws) {
    const int tid = threadIdx.x;
    if (tid < 64) ws[WS_STAT + tid] = 0.0f;
    for (int i = tid; i < OUTC*KPAD; i += blockDim.x) {
        const int row = i / KPAD, k = i % KPAD;
        ws[WS_GWP + i] = (k < KDIM) ? gw[row*KDIM + k] : 0.0f;
    }
}

// One thread per (b,g). Context set staged in LDS; 512-deep inner loop with
// 9 v_exp_f32 per iteration (transcendental-bound phase).
__global__ void k_rbf(const float* __restrict__ xc, const float* __restrict__ yc,
                      const float* __restrict__ xg, const float* __restrict__ sigma,
                      const float* __restrict__ mu, const float* __restrict__ eps1,
                      const float* __restrict__ bu, const float* __restrict__ rw,
                      float* __restrict__ out, float* __restrict__ ws) {
    __shared__ float s_xc[NC*CH];
    __shared__ float s_yc[NC*CH];
    const int idx = blockIdx.x * blockDim.x + threadIdx.x;   // b*NG + g
    const int b = idx / NG;
    const int g = idx % NG;
    // NG % blockDim == 0 -> whole block shares one batch b
    for (int i = threadIdx.x; i < NC*CH; i += blockDim.x) {
        s_xc[i] = xc[b*NC*CH + i];
        s_yc[i] = yc[b*NC*CH + i];
    }
    __syncthreads();

    float invs[NPAR], wmu[NPAR];
#pragma unroll
    for (int p = 0; p < NPAR; ++p) {
        invs[p] = 1.0f / (__expf(sigma[p]) + EPSV);   // 1/(scale+eps) == w_std
        wmu[p]  = __expf(mu[p]);
    }
    float xgv[CH];
#pragma unroll
    for (int c = 0; c < CH; ++c) xgv[c] = xg[(size_t)idx*CH + c];

    float h0[CH][NPAR] = {}, h1[CH][NPAR] = {};
    for (int n = 0; n < NC; ++n) {
#pragma unroll
        for (int c = 0; c < CH; ++c) {
            const float dd = s_xc[n*CH + c] - xgv[c];
            const float yv = s_yc[n*CH + c];
#pragma unroll
            for (int p = 0; p < NPAR; ++p) {
                const float d = dd * invs[p];
                const float w = __expf(-0.5f * d * d);
                h0[c][p] += w;
                h1[c][p] += w * yv;
            }
        }
    }

    // random Fourier prior
    float fp[CH][NPAR] = {};
    for (int k = 0; k < NBAS; ++k) {
        const float rwk = rw[k];
#pragma unroll
        for (int p = 0; p < NPAR; ++p) {
            const float sw = wmu[p] + invs[p] * eps1[(b*NBAS + k)*NPAR + p];
            const float sb = 6.283185307179586f * bu[(b*NBAS + k)*NPAR + p];
#pragma unroll
            for (int c = 0; c < CH; ++c)
                fp[c][p] += rwk * cosf(sw * xgv[c] + sb);
        }
    }
    const float fscale = 0.44721359549995793f;  // sqrt(2/10)

    float* feat = ws + WS_FEAT + (size_t)idx * KPAD;
#pragma unroll
    for (int c = 0; c < CH; ++c) {
#pragma unroll
        for (int p = 0; p < NPAR; ++p) {
            const float fpv  = fscale * fp[c][p];
            const float nh1  = h1[c][p] / (h0[c][p] + EPSV);
            out[O_FP  + ((size_t)idx*CH + c)*NPAR + p] = fpv;
            out[O_NH1 + ((size_t)idx*CH + c)*NPAR + p] = nh1;
            out[O_H0F + (size_t)idx*9 + c*NPAR + p]    = h0[c][p];
            feat[c*NPAR + p] = h0[c][p];                     // feat cols 0..8
            ws[WS_PRE + (((size_t)(p*CH + c)*NBATCH + b)*NG) + g] = nh1 + fpv;
        }
    }
    feat[18] = 0.0f; feat[19] = 0.0f;                        // K padding
}

// Depthwise conv along g + per-(p,c) sum/sumsq via block reduce + atomics.
__global__ void k_conv(const float* __restrict__ w1, const float* __restrict__ b1,
                       const float* __restrict__ w2, const float* __restrict__ b2,
                       const float* __restrict__ w3, const float* __restrict__ b3,
                       float* __restrict__ ws) {
    const int pcb = blockIdx.x;      // pc*NBATCH + b
    const int bb  = pcb % NBATCH;
    const int pc  = pcb / NBATCH;
    const int c   = pc % CH;
    const int p   = pc / CH;
    const float* pre = ws + WS_PRE + (size_t)(pc*NBATCH + bb) * NG;
    float*       z   = ws + WS_Z   + (size_t)(pc*NBATCH + bb) * NG;

    const float* w; const float* bias; int ksz, pad;
    if (p == 0)      { w = w1; bias = b1; ksz = 3; pad = 1; }
    else if (p == 1) { w = w2; bias = b2; ksz = 5; pad = 2; }
    else             { w = w3; bias = b3; ksz = 9; pad = 4; }

    float wloc[9];
    for (int t = 0; t < 9; ++t) wloc[t] = (t < ksz) ? w[c*ksz + t] : 0.0f;
    const float bv = bias[c];

    float lsum = 0.0f, lsq = 0.0f;
    for (int g = threadIdx.x; g < NG; g += blockDim.x) {
        float acc = bv;
        for (int t = 0; t < 9; ++t) {
            const int gi = g + t - pad;
            if (t < ksz && gi >= 0 && gi < NG) acc += wloc[t] * pre[gi];
        }
        z[g] = acc;
        lsum += acc; lsq += acc * acc;
    }

    __shared__ float rs[256], rq[256];
    rs[threadIdx.x] = lsum; rq[threadIdx.x] = lsq;
    __syncthreads();
    for (int s = blockDim.x >> 1; s > 0; s >>= 1) {
        if (threadIdx.x < (unsigned)s) {
            rs[threadIdx.x] += rs[threadIdx.x + s];
            rq[threadIdx.x] += rq[threadIdx.x + s];
        }
        __syncthreads();
    }
    if (threadIdx.x == 0) {
        atomicAdd(ws + WS_STAT + pc*2 + 0, rs[0]);
        atomicAdd(ws + WS_STAT + pc*2 + 1, rq[0]);
    }
}

// BN normalize (training-mode, biased var) + write n_f output + feat cols 9..17.
__global__ void k_bn(const float* __restrict__ gamma, const float* __restrict__ beta,
                     float* __restrict__ out, float* __restrict__ ws) {
    const int idx = blockIdx.x * blockDim.x + threadIdx.x;   // (pc*NBATCH+b)*NG+g
    if (idx >= NPAR*CH*NBATCH*NG) return;
    const int g  = idx % NG;
    const int t  = idx / NG;
    const int bb = t % NBATCH;
    const int pc = t / NBATCH;
    const int c  = pc % CH;
    const int p  = pc / CH;

    const float inv_n = 1.0f / (float)(NBATCH*NG);
    const float m = ws[WS_STAT + pc*2 + 0] * inv_n;
    const float v = ws[WS_STAT + pc*2 + 1] * inv_n - m*m;
    const float zv = ws[WS_Z + idx];
    const float val = gamma[p*CH + c] * (zv - m) * rsqrtf(v + BN_EPSV) + beta[p*CH + c];

    const int bg = bb*NG + g;
    out[O_NF + (size_t)bg*9 + c*NPAR + p] = val;
    ws[WS_FEAT + (size_t)bg*KPAD + 9 + c*NPAR + p] = val;
}

// Final GEMM: y[bg,o] = sum_k feat[bg,k]*g_w[o,k] + g_b[o] via V_WMMA_F32_16X16X4_F32.
// One wave per 16x16 output tile; K=18 padded to 20 -> 5 WMMA steps, pure f32.
// Both A (feat) and B (padded g_w) are KPAD-strided -> unconditional b64 loads.
__global__ void k_gemm(const float* __restrict__ gb, float* __restrict__ out,
                       const float* __restrict__ ws) {
    const int lane = threadIdx.x;        // 0..31 (wave32)
    const int mt   = blockIdx.x;         // 0..511
    const int nt   = blockIdx.y;         // 0..3
    const int rm   = lane & 15;
    const int half = lane >> 4;
    const float* feat = ws + WS_FEAT;
    const float* gwp  = ws + WS_GWP;
    const int ncol = nt*16 + rm;

    v8f acc = {};
#pragma unroll
    for (int s = 0; s < 5; ++s) {
        const int kb = 4*s + 2*half;     // A 16x4 f32 layout: lanes<16 -> K0,1; lanes>=16 -> K2,3
        v2f a, bf;
        a.x  = feat[(size_t)(mt*16 + rm)*KPAD + kb];   // pad cols 18/19 are zero
        a.y  = feat[(size_t)(mt*16 + rm)*KPAD + kb + 1];
        bf.x = gwp[(size_t)ncol*KPAD + kb];            // B[k][n] = g_w[n][k], zero-padded
        bf.y = gwp[(size_t)ncol*KPAD + kb + 1];
        acc = __builtin_amdgcn_wmma_f32_16x16x4_f32(
            false, a, false, bf, (short)0, acc, false, false);
    }
    const float bias = gb[ncol];
#pragma unroll
    for (int v = 0; v < 8; ++v) {        // C layout: VGPR v -> row v + 8*half, col = lane&15
        const int mrow = mt*16 + half*8 + v;
        out[(size_t)mrow*OUTC + ncol] = acc[v] + bias;
    }
}

extern "C" void kernel_launch(void* const* d_in, const int* in_sizes, int n_in,
                              void* d_out, int out_size, void* d_ws, size_t ws_size,
                              hipStream_t stream) {
    (void)in_sizes; (void)n_in; (void)out_size; (void)ws_size;
    const float* xc    = (const float*)d_in[0];
    const float* yc    = (const float*)d_in[1];
    const float* xg    = (const float*)d_in[2];
    const float* sigma = (const float*)d_in[3];
    const float* mu    = (const float*)d_in[4];
    const float* eps1  = (const float*)d_in[5];
    const float* bu    = (const float*)d_in[6];
    const float* rw    = (const float*)d_in[7];
    const float* cw1   = (const float*)d_in[8];
    const float* cb1   = (const float*)d_in[9];
    const float* cw2   = (const float*)d_in[10];
    const float* cb2   = (const float*)d_in[11];
    const float* cw3   = (const float*)d_in[12];
    const float* cb3   = (const float*)d_in[13];
    const float* gamma = (const float*)d_in[14];
    const float* beta  = (const float*)d_in[15];
    const float* gw    = (const float*)d_in[16];
    const float* gb    = (const float*)d_in[17];
    float* out = (float*)d_out;
    float* ws  = (float*)d_ws;

    k_setup<<<1, 256, 0, stream>>>(gw, ws);
    k_rbf<<<MROWS/256, 256, 0, stream>>>(xc, yc, xg, sigma, mu, eps1, bu, rw, out, ws);
    k_conv<<<NPAR*CH*NBATCH, 256, 0, stream>>>(cw1, cb1, cw2, cb2, cw3, cb3, ws);
    k_bn<<<(NPAR*CH*NBATCH*NG + 255)/256, 256, 0, stream>>>(gamma, beta, out, ws);
    k_gemm<<<dim3(MROWS/16, OUTC/16), 32, 0, stream>>>(gb, out, ws);
}